// Vlasov_51170240364848
// MI455X (gfx1250) — compile-verified
//
#include <hip/hip_runtime.h>
#include <stdint.h>
#include <stddef.h>

// ---------------------------------------------------------------------------
// Vlasov spherical-harmonic update, MI455X (gfx1250, wave32).
// Bandwidth-bound (~330 MB @ 23.3 TB/s => ~15us floor). fp32 throughout.
// CDNA5 paths:
//   * async global->LDS staging (ASYNCcnt + s_wait_asynccnt)
//   * E+B harmonic coupling cast as exact per-(x,y) GEMM, executed with
//     V_WMMA_F32_16X16X4_F32 (D[32x192] = M[32x64] x Q[64x192])
//   * nontemporal streaming stores for the write-once output.
// ---------------------------------------------------------------------------

typedef float v2f __attribute__((ext_vector_type(2)));
typedef float v8f __attribute__((ext_vector_type(8)));

namespace {

constexpr int   kNL = 3;
constexpr int   kL  = 4;          // NL + 1
constexpr int   kNX = 128;
constexpr int   kNY = 64;
constexpr int   kNV = 192;
constexpr float kDX = 0.1f;
constexpr float kDY = 0.1f;
constexpr float kDV = 0.05f;
constexpr int   kNH = 10;         // harmonics with im <= il

constexpr int QP = 200;           // Q LDS pitch (200 mod 64 = 8 -> lane halves 16 banks apart)
constexpr int DP = 196;           // D LDS pitch (8*196 mod 64 = 32 -> conflict-free tile writes)
constexpr int MP = 68;            // M LDS pitch (68 mod 64 = 4 -> conflict-free A reads)

__host__ __device__ constexpr int H(int il, int im) { return il * (il + 1) / 2 + im; }
__host__ __device__ constexpr int FL(int il, int im) { return il * kL + im; }

// Coefficient tables (match _edfdv_coeffs incl. overrides).
__host__ __device__ constexpr float fA1(int il, int im) {
  return (il == 0 && im == 0) ? 1.0f
       : ((float)(il + 1 - im) / (float)(2 * il + 1)) * ((float)il / (2.0f * (float)(il + 1)));
}
__host__ __device__ constexpr float fC1(int il) {
  return (il == 0) ? 0.5f
       : (1.0f / (float)(2 * il + 1)) * ((float)il / (2.0f * (float)(il + 1)));
}
__host__ __device__ constexpr float fC2(int il, int im) {
  return -((float)((il - im + 2) * (il - im + 1)) / (float)(2 * il + 1))
         * ((float)il / (2.0f * (float)(il + 1)));
}
__host__ __device__ constexpr float fB1(int il) {
  return -((float)il / (float)(2 * il + 1)) * ((float)il / 2.0f);
}
__host__ __device__ constexpr float fA2(int il, int im) {
  return (il == 1 && im == 0) ? (1.0f / 3.0f) : (float)(il + im) / (float)(2 * il + 1);
}
__host__ __device__ constexpr float fC3(int il, int im) {
  return (il == 1 && im == 1) ? -0.1f : 0.5f / (float)(2 * il + 1);
}
__host__ __device__ constexpr float fC4(int il, int im) {
  return -0.5f * (float)((il + im - 1) * (il + im)) / (float)(2 * il + 1);
}

__device__ __forceinline__ void rfma(float& ar, float& ai, float c, float sr, float si) {
  ar += c * sr;
  ai += c * si;
}

} // namespace

// Q row index for (harmonic k, quantity q, part p); q: 0=g, 1=h, 2=f.
#define QROW(k, q, p) (6 * (k) + 2 * (q) + (p))

__global__ void __launch_bounds__(kNV)
vlasov_cdna5_kernel(const float* __restrict__ f_real,
                    const float* __restrict__ f_imag,
                    const float* __restrict__ e,
                    const float* __restrict__ b,
                    const float* __restrict__ vgrid,
                    float* __restrict__ out) {
  const int x = blockIdx.x;
  const int y = blockIdx.y;
  const int v = threadIdx.x;          // one thread per velocity cell
  const int lane = threadIdx.x & 31;
  const int wid  = threadIdx.x >> 5;  // 0..5 (6 wave32s)

  // Q: 60 used rows (g/h/f re+im per harmonic) + 4 zero pad rows, cols = v.
  // D: E+B coupling result, 20 used rows. M: 20x60 coupling matrix (padded).
  __shared__ float sQ[64][QP];
  __shared__ float sD[32][DP];
  __shared__ float sM[32][MP];

  // ---- Phase 0: async-stage center f columns straight into Q's f rows -----
  {
    const float* bases[2] = { f_real, f_imag };
#pragma unroll
    for (int il = 0; il < kL; ++il) {
#pragma unroll
      for (int im = 0; im <= il; ++im) {
        const int k = H(il, im);
        const size_t goff =
            ((((size_t)FL(il, im) * kNX + (size_t)x) * kNY + (size_t)y) * kNV) + (size_t)v;
#pragma unroll
        for (int p = 0; p < 2; ++p) {
          const float* gp = bases[p] + goff;
          unsigned lds_addr = (unsigned)(size_t)(&sQ[QROW(k, 2, p)][v]);
          asm volatile("global_load_async_to_lds_b32 %0, %1, off"
                       :: "v"(lds_addr), "v"(gp)
                       : "memory");
        }
      }
    }
  }
  // Cooperative zero of M and Q pad rows while the async engine runs.
  for (int i = threadIdx.x; i < 32 * MP; i += kNV) (&sM[0][0])[i] = 0.0f;
  for (int i = threadIdx.x; i < 4 * QP; i += kNV) (&sQ[60][0])[i] = 0.0f;

  asm volatile("s_wait_asynccnt 0" ::: "memory");
  __syncthreads();

  // ---- Phase 1: per-thread g,h from the staged f columns ------------------
  const float vv   = vgrid[v];
  const float invv = 1.0f / vv;
#pragma unroll
  for (int il = 0; il < kL; ++il) {
#pragma unroll
    for (int im = 0; im <= il; ++im) {
      const int k = H(il, im);
      const float cr = sQ[QROW(k, 2, 0)][v];
      const float ci = sQ[QROW(k, 2, 1)][v];
      // pad = [-f0, f.., 0]; dfdv = (pad[v+2]-pad[v]) / (2*DV)
      const float nr = (v < kNV - 1) ? sQ[QROW(k, 2, 0)][v + 1] : 0.0f;
      const float ni = (v < kNV - 1) ? sQ[QROW(k, 2, 1)][v + 1] : 0.0f;
      const float pr = (v > 0) ? sQ[QROW(k, 2, 0)][v - 1] : -cr;
      const float pi = (v > 0) ? sQ[QROW(k, 2, 1)][v - 1] : -ci;
      const float dvr = (nr - pr) * (0.5f / kDV);
      const float dvi = (ni - pi) * (0.5f / kDV);
      const float gl = (float)il * invv;
      const float hl = (float)(il + 1) * invv;
      sQ[QROW(k, 0, 0)][v] = dvr - gl * cr;
      sQ[QROW(k, 0, 1)][v] = dvi - gl * ci;
      sQ[QROW(k, 1, 0)][v] = hl * cr + dvr;
      sQ[QROW(k, 1, 1)][v] = hl * ci + dvi;
    }
  }
  __syncthreads();   // M zeroed + Q complete

  // ---- Phase 2: thread 0 builds the exact E+B coupling matrix M -----------
  // Transcribes the reference block order (incl. the two overwrite
  // assignments, which reset whole rows) with matrix-row accumulators.
  if (threadIdx.x == 0) {
    const size_t xy3 = ((size_t)x * kNY + (size_t)y) * 3;
    const float Ex = e[xy3 + 0], E1 = e[xy3 + 1], E2 = e[xy3 + 2];
    const float Bx = b[xy3 + 0], B1 = b[xy3 + 1], B2 = b[xy3 + 2];
    // ep=(E1,E2) em=(E1,-E2) bm=(B1,B2) bp=(B1,-B2)

    auto CTERM = [&](int t, float c, float wr, float wi, int s, int q) {
      // delta[t] += c*(wr+i*wi)*(quantity q of harmonic s)
      sM[2 * t][QROW(s, q, 0)]     += c * wr;
      sM[2 * t][QROW(s, q, 1)]     -= c * wi;
      sM[2 * t + 1][QROW(s, q, 0)] += c * wi;
      sM[2 * t + 1][QROW(s, q, 1)] += c * wr;
    };
    auto RTERM = [&](int t, float c, int s, int q) {
      sM[2 * t][QROW(s, q, 0)]     += c;
      sM[2 * t + 1][QROW(s, q, 1)] += c;
    };
    auto ZROW = [&](int t) {
      for (int c = 0; c < 64; ++c) { sM[2 * t][c] = 0.0f; sM[2 * t + 1][c] = 0.0f; }
    };

    // E1: a1*ex*g[il][0] (reference quirk) and c1*em*g
    for (int il = 0; il < kNL; ++il) {
      for (int im = 0; im <= il; ++im)
        RTERM(H(il + 1, im), fA1(il, im) * Ex, H(il, 0), 0);
      for (int im = 0; im < il; ++im)
        CTERM(H(il + 1, im + 1), fC1(il), E1, -E2, H(il, im), 0);
    }
    // E2: c2*ep*g; then OVERWRITE delta[il+1][0] = b1*Re(ep*g[il][1])
    for (int il = 1; il < kNL; ++il) {
      for (int im = 1; im <= il; ++im)
        CTERM(H(il + 1, im - 1), fC2(il, im), E1, E2, H(il, im), 0);
      const int t = H(il + 1, 0);
      ZROW(t);
      sM[2 * t][QROW(H(il, 1), 0, 0)] = fB1(il) * E1;   // Re part only,
      sM[2 * t][QROW(H(il, 1), 0, 1)] = -fB1(il) * E2;  // imag row stays zero
    }
    // E3: a2*ex*h and c4*ep*h
    for (int il = 1; il <= kNL; ++il) {
      for (int im = 0; im < il; ++im)
        RTERM(H(il - 1, im), fA2(il, im) * Ex, H(il, im), 1);
      for (int im = 1; im <= il; ++im)
        CTERM(H(il - 1, im - 1), fC4(il, im), E1, E2, H(il, im), 1);
    }
    // E4: c3*em*h; plus repeated c4[il][1]*ep*h[il][1] (reference repeats it)
    for (int il = 2; il <= kNL; ++il)
      for (int im = 0; im < il; ++im)
        CTERM(H(il - 1, im + 1), fC3(il, im), E1, -E2, H(il, im), 1);
    for (int il = 1; il <= kNL; ++il)
      CTERM(H(il - 1, 0), fC4(il, 1), E1, E2, H(il, 1), 1);
    // B block: OVERWRITE delta[il][im+1] = 0.5*bm*f[il][im]; then additions
    for (int il = 1; il <= kNL; ++il) {
      for (int im = 0; im < il; ++im) {
        const int t = H(il, im + 1);
        ZROW(t);
        CTERM(t, 0.5f, B1, B2, H(il, im), 2);
      }
      for (int im = 1; im <= il; ++im) {
        const int s = H(il, im);
        const float mb = (float)im * Bx;              // -1j*im*bx*f
        sM[2 * s][QROW(s, 2, 1)]     += mb;
        sM[2 * s + 1][QROW(s, 2, 0)] -= mb;
        CTERM(H(il, im - 1), -0.5f * (float)((il - im + 1) * (il + im)),
              B1, B2, s, 2);
      }
      // delta[il][0] += Re(-il*(il+1)*bp*f[il][1]),  bp=(B1,-B2)
      const float c = (float)(-il * (il + 1));
      sM[2 * H(il, 0)][QROW(H(il, 1), 2, 0)] += c * B1;
      sM[2 * H(il, 0)][QROW(H(il, 1), 2, 1)] += c * B2;
    }
  }
  __syncthreads();

  // ---- Phase 3: D[32x192] = M[32x64] x Q[64x192] via WMMA f32 16x16x4 -----
  {
    const int mt  = wid / 3;            // 2 row-tiles
    const int nt0 = (wid % 3) * 4;      // 12 col-tiles, 4 per wave
    const int lr  = lane & 15;
    const int hi2 = (lane >> 4) << 1;   // lanes 16-31 hold K+2
#pragma unroll
    for (int j = 0; j < 4; ++j) {
      const int colB = (nt0 + j) * 16 + lr;
      v8f c = {};
#pragma unroll
      for (int kk = 0; kk < 16; ++kk) {
        v2f a, bb;
        a.x  = sM[mt * 16 + lr][kk * 4 + hi2];
        a.y  = sM[mt * 16 + lr][kk * 4 + hi2 + 1];
        bb.x = sQ[kk * 4 + hi2][colB];
        bb.y = sQ[kk * 4 + hi2 + 1][colB];
        c = __builtin_amdgcn_wmma_f32_16x16x4_f32(
                false, a, false, bb, (short)0, c, false, false);
      }
      const int rbase = mt * 16 + ((lane >= 16) ? 8 : 0);
#pragma unroll
      for (int r = 0; r < 8; ++r)
        sD[rbase + r][colB] = c[r];
    }
  }
  __syncthreads();

  // ---- Phase 4: pick up E+B result, add spatial S-terms, stream out -------
  float dr[kNH], di[kNH];
#pragma unroll
  for (int k = 0; k < kNH; ++k) { dr[k] = sD[2 * k][v]; di[k] = sD[2 * k + 1][v]; }

  const int xp = (x + 1) & (kNX - 1);
  const int xm = (x - 1) & (kNX - 1);
  const int yp = (y < kNY - 1) ? (y + 1) : y;
  const int ym = (y > 0) ? (y - 1) : y;
  const float yscale = (y == 0 || y == kNY - 1) ? (1.0f / kDY) : (0.5f / kDY);

#pragma unroll
  for (int il = 0; il <= kNL; ++il) {
#pragma unroll
    for (int im = 0; im <= il; ++im) {
      const size_t fbase = (size_t)FL(il, im) * kNX;
      const size_t oxp = ((fbase + (size_t)xp) * kNY + (size_t)y) * kNV + (size_t)v;
      const size_t oxm = ((fbase + (size_t)xm) * kNY + (size_t)y) * kNV + (size_t)v;
      const size_t oyp = ((fbase + (size_t)x) * kNY + (size_t)yp) * kNV + (size_t)v;
      const size_t oym = ((fbase + (size_t)x) * kNY + (size_t)ym) * kNV + (size_t)v;
      const float dxr = vv * (f_real[oxp] - f_real[oxm]) * (0.5f / kDX);
      const float dxi = vv * (f_imag[oxp] - f_imag[oxm]) * (0.5f / kDX);
      const float dyr = vv * (f_real[oyp] - f_real[oym]) * yscale;
      const float dyi = vv * (f_imag[oyp] - f_imag[oym]) * yscale;

      // S1 (ddx): il -> il+1, same im
      if (il < kNL)
        rfma(dr[H(il + 1, im)], di[H(il + 1, im)],
             -((float)(il - im + 1) / (float)(2 * il + 1)), dxr, dxi);
      // S2 (ddx): il -> il-1, same im (il > im)
      if (il >= 1 && im < il)
        rfma(dr[H(il - 1, im)], di[H(il - 1, im)],
             -((float)(il + im) / (float)(2 * il + 1)), dxr, dxi);
      // S1 (ddy): source (il,1), il in [1,NL) -> (il+1,0)
      if (im == 1 && il >= 1 && il < kNL)
        rfma(dr[H(il + 1, 0)], di[H(il + 1, 0)],
             (float)(il * (il + 1)) / (float)(2 * il + 1), dyr, dyi);
      // S2 (ddy): source (il,1), il in [1,NL] -> (il-1,0)
      if (im == 1 && il >= 1)
        rfma(dr[H(il - 1, 0)], di[H(il - 1, 0)],
             (float)(il * (il + 1)) / (float)(2 * il + 1), dyr, dyi);
      // S3 part A: il in [1,NL)
      if (il >= 1 && il < kNL) {
        if (im >= 1 && im < il)
          rfma(dr[H(il + 1, im + 1)], di[H(il + 1, im + 1)],
               -0.5f / (float)(2 * il + 1), dyr, dyi);
        if (im >= 1)
          rfma(dr[H(il + 1, im - 1)], di[H(il + 1, im - 1)],
               0.5f * (float)((il - im + 2) * (il - im + 1)) / (float)(2 * il + 1),
               dyr, dyi);
      }
      // S3 part B: il in [2,NL]
      if (il >= 2) {
        if (im < il)
          rfma(dr[H(il - 1, im + 1)], di[H(il - 1, im + 1)],
               0.5f / (float)(2 * il + 1), dyr, dyi);
        if (im >= 1)
          rfma(dr[H(il - 1, im - 1)], di[H(il - 1, im - 1)],
               -0.5f * (float)((il + im - 1) * (il + im)) / (float)(2 * il + 1),
               dyr, dyi);
      }
    }
  }

  // Streaming (nontemporal) stores of (2, L, L, NX, NY, NV) output.
  const size_t imag_off = (size_t)kL * kL * kNX * kNY * kNV;
#pragma unroll
  for (int il = 0; il < kL; ++il) {
#pragma unroll
    for (int im = 0; im < kL; ++im) {
      const size_t oo =
          ((((size_t)FL(il, im)) * kNX + (size_t)x) * kNY + (size_t)y) * kNV + (size_t)v;
      float vr = 0.0f, vi = 0.0f;
      if (im <= il) { vr = dr[H(il, im)]; vi = di[H(il, im)]; }
      __builtin_nontemporal_store(vr, out + oo);
      __builtin_nontemporal_store(vi, out + imag_off + oo);
    }
  }
}

extern "C" void kernel_launch(void* const* d_in, const int* in_sizes, int n_in,
                              void* d_out, int out_size, void* d_ws, size_t ws_size,
                              hipStream_t stream) {
  (void)in_sizes; (void)n_in; (void)out_size; (void)d_ws; (void)ws_size;
  const float* f_real = (const float*)d_in[0];
  const float* f_imag = (const float*)d_in[1];
  const float* e      = (const float*)d_in[2];
  const float* b      = (const float*)d_in[3];
  const float* vgrid  = (const float*)d_in[4];
  float* out = (float*)d_out;

  dim3 grid(kNX, kNY);           // one block per (x,y)
  dim3 block(kNV);               // 192 threads = 6 wave32s
  vlasov_cdna5_kernel<<<grid, block, 0, stream>>>(f_real, f_imag, e, b, vgrid, out);
}